// YOLOv3_87290915324227
// MI455X (gfx1250) — compile-verified
//
#include <hip/hip_runtime.h>
#include <math.h>

#define NCLS 80
#define NBINS 4096
#define TOPK_ 100
#define CAP 8192
#define CONF_TH_ 0.001f
#define NMS_TH_ 0.5f
#define INPUT_SZ_ 2048.0f

typedef float v2f __attribute__((ext_vector_type(2)));
typedef float v8f __attribute__((ext_vector_type(8)));

// fast exp for the bulk scoring paths (native v_exp_f32); K1 and K3 must use
// the SAME function so histogram bins and compaction threshold agree.
__device__ __forceinline__ float fexp(float x) { return __expf(x); }

__device__ __forceinline__ float sigm(float x) { return 1.0f / (1.0f + expf(-x)); }
// strict "a precedes b" in (score desc, index asc) order
__device__ __forceinline__ bool prec(float sa, int ia, float sb, int ib) {
  return (sa > sb) || (sa == sb && ia < ib);
}

// ---------------------------------------------------------------- init
__global__ __launch_bounds__(256) void init_kernel(unsigned* __restrict__ p, int n) {
  int t = blockIdx.x * 256 + threadIdx.x;
  if (t < n) p[t] = 0u;
}

// -------------------------------------------------- pass 1: prob + histogram
// One wave handles 16 anchor rows (80 classes each). Lane L and lane L+16
// both work on row (L&15): lane<16 covers classes 4i,4i+1; lane>=16 covers
// 4i+2,4i+3 -- exactly the A-matrix layout of V_WMMA_F32_16X16X4_F32, so the
// softmax denominator is 20 accumulating WMMAs against an all-ones B.
__global__ __launch_bounds__(256) void score_hist_kernel(
    const float* __restrict__ conf, const float* __restrict__ cls,
    float* __restrict__ rowA, float* __restrict__ rowM,
    unsigned* __restrict__ hist) {
  __shared__ unsigned shist[NBINS];
  __shared__ float ssum[8 * 16];
  const int tid = threadIdx.x;
  for (int b = tid; b < NBINS; b += 256) shist[b] = 0u;

  const int wid = tid >> 5;
  const int lane = tid & 31;
  const int half = lane >> 4;
  const int lrow = lane & 15;
  const int r = (blockIdx.x * 8 + wid) * 16 + lrow;
  const float* __restrict__ crow = cls + (size_t)r * NCLS;
  __builtin_prefetch(crow, 0, 1);  // global_prefetch_b8

  __syncthreads();

  // pass A: row max (each lane covers 40 of 80, merge across half-waves)
  float m = -3.402823466e38f;
#pragma unroll
  for (int i = 0; i < 20; ++i) {
    float2 v = *(const float2*)(crow + 4 * i + 2 * half);
    m = fmaxf(m, fmaxf(v.x, v.y));
  }
  m = fmaxf(m, __shfl_xor(m, 16, 32));

  // pass B: sum(exp(x-m)) per row on the matrix pipe
  v8f acc = {0.f, 0.f, 0.f, 0.f, 0.f, 0.f, 0.f, 0.f};
  v2f bones;
  bones.x = 1.0f;
  bones.y = 1.0f;
#pragma unroll 4
  for (int i = 0; i < 20; ++i) {
    float2 v = *(const float2*)(crow + 4 * i + 2 * half);
    v2f a;
    a.x = fexp(v.x - m);
    a.y = fexp(v.y - m);
    acc = __builtin_amdgcn_wmma_f32_16x16x4_f32(false, a, false, bones,
                                                (short)0, acc, false, false);
  }
  // D layout: lanes 0-15 VGPR g = rowsum(g); lanes 16-31 VGPR g = rowsum(g+8)
  if (lane == 0) {
#pragma unroll
    for (int g = 0; g < 8; ++g) ssum[wid * 16 + g] = acc[g];
  } else if (lane == 16) {
#pragma unroll
    for (int g = 0; g < 8; ++g) ssum[wid * 16 + 8 + g] = acc[g];
  }
  __syncthreads();
  const float rsum = ssum[wid * 16 + lrow];
  const float A = sigm(conf[r]) / rsum;
  if (half == 0) {
    rowA[r] = A;
    rowM[r] = m;
  }

  // pass C: probs -> block-local histogram
#pragma unroll 4
  for (int i = 0; i < 20; ++i) {
    float2 v = *(const float2*)(crow + 4 * i + 2 * half);
    float p0 = A * fexp(v.x - m);
    float p1 = A * fexp(v.y - m);
    int b0 = min(NBINS - 1, max(0, (int)(p0 * (float)NBINS)));
    int b1 = min(NBINS - 1, max(0, (int)(p1 * (float)NBINS)));
    atomicAdd(&shist[b0], 1u);
    atomicAdd(&shist[b1], 1u);
  }
  __syncthreads();
  for (int b = tid; b < NBINS; b += 256)
    if (shist[b]) atomicAdd(&hist[b], shist[b]);
}

// -------------------------------------------------- pass 2: pick threshold
__global__ void thresh_kernel(const unsigned* __restrict__ hist,
                              float* __restrict__ thr) {
  int l = threadIdx.x;
  if (l < 3) {
    const unsigned* h = hist + l * NBINS;
    unsigned accn = 0;
    int b = 0;
    for (int i = NBINS - 1; i >= 0; --i) {
      accn += h[i];
      if (accn >= TOPK_) { b = i; break; }
    }
    int bs = (b > 0) ? b - 1 : 0;  // one-bin safety margin vs float truncation
    thr[l] = (float)bs / (float)NBINS;
  }
}

// -------------------------------------------------- pass 3: compact candidates
__global__ __launch_bounds__(256) void compact_kernel(
    const float* __restrict__ cls, const float* __restrict__ rowA,
    const float* __restrict__ rowM, const float* __restrict__ thr,
    unsigned* __restrict__ cnt, float2* __restrict__ cand) {
  const int tid = threadIdx.x;
  const int wid = tid >> 5;
  const int lane = tid & 31;
  const int r = blockIdx.x * 16 + wid * 2 + (lane >> 4);
  const int c0 = lane & 15;
  const float th = *thr;
  const float A = rowA[r];
  const float m = rowM[r];
  const float* __restrict__ crow = cls + (size_t)r * NCLS;
  __builtin_prefetch(crow, 0, 1);
#pragma unroll
  for (int c = c0; c < NCLS; c += 16) {
    float p = A * fexp(crow[c] - m);
    if (p >= th) {
      unsigned pos = atomicAdd(cnt, 1u);
      if (pos < CAP) {
        float2 e;
        e.x = p;
        e.y = __int_as_float(r * NCLS + c);
        cand[pos] = e;
      }
    }
  }
}

// -------------------------------------------------- pass 4: per-level top-100
__global__ __launch_bounds__(1024) void topk_kernel(
    const unsigned* __restrict__ cnt, const float2* __restrict__ cand,
    const float* __restrict__ reg, const float* __restrict__ anchors,
    float stridef, float* __restrict__ stage) {
  __shared__ float ss[CAP];
  __shared__ int si[CAP];
  const int tid = threadIdx.x;
  const int n = min((int)*cnt, CAP);
  for (int t = tid; t < CAP; t += 1024) {
    if (t < n) {
      float2 e = cand[t];
      ss[t] = e.x;
      si[t] = __float_as_int(e.y);
    } else {
      ss[t] = -3.402823466e38f;
      si[t] = 0x7fffffff;
    }
  }
  __syncthreads();
  for (int k = 2; k <= CAP; k <<= 1) {
    for (int j = k >> 1; j > 0; j >>= 1) {
      for (int t = tid; t < CAP; t += 1024) {
        int x = t ^ j;
        if (x > t) {
          bool dir = ((t & k) == 0);
          float s1 = ss[t], s2 = ss[x];
          int i1 = si[t], i2 = si[x];
          bool sw = dir ? prec(s2, i2, s1, i1) : prec(s1, i1, s2, i2);
          if (sw) {
            ss[t] = s2; ss[x] = s1;
            si[t] = i2; si[x] = i1;
          }
        }
      }
      __syncthreads();
    }
  }
  if (tid < TOPK_) {
    float s = ss[tid];
    int idx = si[tid];
    float o0 = 0.f, o1 = 0.f, o2 = 0.f, o3 = 0.f, o4 = 0.f, o5 = 0.f;
    if (idx != 0x7fffffff) {
      int a = idx / NCLS;
      int lab = idx - a * NCLS;
      float r0 = reg[a * 4 + 0], r1 = reg[a * 4 + 1];
      float r2 = reg[a * 4 + 2], r3 = reg[a * 4 + 3];
      float a0 = anchors[a * 4 + 0], a1 = anchors[a * 4 + 1];
      float a2 = anchors[a * 4 + 2], a3 = anchors[a * 4 + 3];
      float cx = (sigm(r0) + a0) * stridef;
      float cy = (sigm(r1) + a1) * stridef;
      float w = expf(r2) * a2;   // precise: lands directly in output boxes
      float h = expf(r3) * a3;
      o0 = s;
      o1 = (float)lab;
      o2 = cx - 0.5f * w;
      o3 = cy - 0.5f * h;
      o4 = cx + 0.5f * w;
      o5 = cy + 0.5f * h;
    }
    float* dst = stage + (size_t)tid * 8;
    dst[0] = o0; dst[1] = o1; dst[2] = o2;
    dst[3] = o3; dst[4] = o4; dst[5] = o5;
  }
}

// -------------------------------------------------- pass 5: sort + NMS + out
__global__ __launch_bounds__(256) void nms_kernel(const float* __restrict__ stage,
                                                  float* __restrict__ out) {
  const int K = 300, M = 512;
  __shared__ float s[M], lab[M], bx1[M], by1[M], bx2[M], by2[M], area[M];
  __shared__ int oi[M], keep[M];
  const int tid = threadIdx.x;
  for (int t = tid; t < M; t += 256) {
    if (t < K) {
      const float* src = stage + (size_t)t * 8;
      s[t] = src[0]; lab[t] = src[1];
      bx1[t] = src[2]; by1[t] = src[3];
      bx2[t] = src[4]; by2[t] = src[5];
      oi[t] = t;
    } else {
      s[t] = -3.402823466e38f;
      oi[t] = 0x7fffffff;
      lab[t] = 0.f; bx1[t] = 0.f; by1[t] = 0.f; bx2[t] = 0.f; by2[t] = 0.f;
    }
  }
  __syncthreads();
  for (int k = 2; k <= M; k <<= 1) {
    for (int j = k >> 1; j > 0; j >>= 1) {
      for (int t = tid; t < M; t += 256) {
        int x = t ^ j;
        if (x > t) {
          bool dir = ((t & k) == 0);
          bool sw = dir ? prec(s[x], oi[x], s[t], oi[t])
                        : prec(s[t], oi[t], s[x], oi[x]);
          if (sw) {
            float f;
            int ii;
            f = s[t]; s[t] = s[x]; s[x] = f;
            ii = oi[t]; oi[t] = oi[x]; oi[x] = ii;
            f = lab[t]; lab[t] = lab[x]; lab[x] = f;
            f = bx1[t]; bx1[t] = bx1[x]; bx1[x] = f;
            f = by1[t]; by1[t] = by1[x]; by1[x] = f;
            f = bx2[t]; bx2[t] = bx2[x]; bx2[x] = f;
            f = by2[t]; by2[t] = by2[x]; by2[x] = f;
          }
        }
      }
      __syncthreads();
    }
  }
  for (int t = tid; t < K; t += 256) {
    area[t] = (bx2[t] - bx1[t]) * (by2[t] - by1[t]);
    keep[t] = (s[t] > CONF_TH_) ? 1 : 0;
  }
  __syncthreads();
  for (int i = 0; i < K; ++i) {
    if (keep[i]) {
      float li = lab[i], xi1 = bx1[i], yi1 = by1[i];
      float xi2 = bx2[i], yi2 = by2[i], ai = area[i];
      for (int j = i + 1 + tid; j < K; j += 256) {
        if (keep[j] && lab[j] == li) {
          float xx1 = fmaxf(xi1, bx1[j]);
          float yy1 = fmaxf(yi1, by1[j]);
          float xx2 = fminf(xi2, bx2[j]);
          float yy2 = fminf(yi2, by2[j]);
          float w = fmaxf(1e-10f, xx2 - xx1);
          float h = fmaxf(1e-10f, yy2 - yy1);
          float inter = w * h;
          float iou = inter / (ai + area[j] - inter);
          if (iou > NMS_TH_) keep[j] = 0;
        }
      }
    }
    __syncthreads();
  }
  for (int t = tid; t < K; t += 256) {
    float* dst = out + (size_t)t * 6;
    if (keep[t]) {
      dst[0] = fminf(fmaxf(bx1[t] / INPUT_SZ_, 0.f), 1.f);
      dst[1] = fminf(fmaxf(by1[t] / INPUT_SZ_, 0.f), 1.f);
      dst[2] = fminf(fmaxf(bx2[t] / INPUT_SZ_, 0.f), 1.f);
      dst[3] = fminf(fmaxf(by2[t] / INPUT_SZ_, 0.f), 1.f);
      dst[4] = s[t];
      dst[5] = lab[t];
    } else {
      dst[0] = dst[1] = dst[2] = dst[3] = dst[4] = dst[5] = 0.f;
    }
  }
}

// ---------------------------------------------------------------- launch
extern "C" void kernel_launch(void* const* d_in, const int* in_sizes, int n_in,
                              void* d_out, int out_size, void* d_ws, size_t ws_size,
                              hipStream_t stream) {
  (void)in_sizes; (void)n_in; (void)out_size; (void)ws_size;
  unsigned char* ws = (unsigned char*)d_ws;
  // workspace layout (~2.3 MB total)
  unsigned* hist = (unsigned*)(ws + 0);        // 3*4096 u32
  unsigned* cnt = (unsigned*)(ws + 49152);     // 3 u32 (contiguous with hist)
  float* thr = (float*)(ws + 49168);           // 3 f32
  float2* cand = (float2*)(ws + 65536);        // 3*CAP float2
  float* rowA = (float*)(ws + 262144);         // 258048 f32
  float* rowM = (float*)(ws + 1294336);        // 258048 f32
  float* stage = (float*)(ws + 2326528);       // 300*8 f32

  const int R[3] = {196608, 49152, 12288};
  const int rowOff[3] = {0, 196608, 245760};
  const float strf[3] = {8.f, 16.f, 32.f};

  // zero histograms + counters (hist and cnt are contiguous u32s)
  init_kernel<<<(3 * NBINS + 3 + 255) / 256, 256, 0, stream>>>(hist, 3 * NBINS + 3);

  for (int l = 0; l < 3; ++l) {
    const float* conf = (const float*)d_in[4 * l + 0];
    const float* cls = (const float*)d_in[4 * l + 1];
    score_hist_kernel<<<R[l] / 128, 256, 0, stream>>>(
        conf, cls, rowA + rowOff[l], rowM + rowOff[l], hist + l * NBINS);
  }
  thresh_kernel<<<1, 32, 0, stream>>>(hist, thr);
  for (int l = 0; l < 3; ++l) {
    const float* cls = (const float*)d_in[4 * l + 1];
    compact_kernel<<<R[l] / 16, 256, 0, stream>>>(
        cls, rowA + rowOff[l], rowM + rowOff[l], thr + l, cnt + l,
        cand + (size_t)l * CAP);
  }
  for (int l = 0; l < 3; ++l) {
    const float* reg = (const float*)d_in[4 * l + 2];
    const float* anc = (const float*)d_in[4 * l + 3];
    topk_kernel<<<1, 1024, 0, stream>>>(cnt + l, cand + (size_t)l * CAP, reg, anc,
                                        strf[l], stage + (size_t)l * TOPK_ * 8);
  }
  nms_kernel<<<1, 256, 0, stream>>>(stage, (float*)d_out);
}